// SemiFLGC_21139829031412
// MI455X (gfx1250) — compile-verified
//
#include <hip/hip_runtime.h>
#include <cstddef>

#define FD 128   // feature dim
#define NC 40    // classes
#define NCP 48   // classes padded to 3 x 16 tiles

typedef __attribute__((ext_vector_type(2))) float v2f;
typedef __attribute__((ext_vector_type(8))) float v8f;
typedef int v4i __attribute__((vector_size(16)));  // matches async-LDS builtin param

// D(16x16,f32) = A(16x4,f32) x B(4x16,f32) + C ; fp32 WMMA, wave32
__device__ __forceinline__ v8f wmma4(v2f a, v2f b, v8f c) {
  return __builtin_amdgcn_wmma_f32_16x16x4_f32(false, a, false, b, (short)0, c,
                                               false, false);
}

// ---------------- small utility kernels ----------------

__global__ void fill_kernel(float* p, float v, int n) {
  int i = blockIdx.x * blockDim.x + threadIdx.x;
  if (i < n) p[i] = v;
}

__global__ void fill_int_kernel(int* p, int v, int n) {
  int i = blockIdx.x * blockDim.x + threadIdx.x;
  if (i < n) p[i] = v;
}

__global__ void copy_int_kernel(const int* __restrict__ src, int* __restrict__ dst, int n) {
  int i = blockIdx.x * blockDim.x + threadIdx.x;
  if (i < n) dst[i] = src[i];
}

// deg[col[e]] += 1  (deg pre-filled with 1.0 for the self loop)
__global__ void deg_kernel(const int* __restrict__ col, float* __restrict__ deg, int E) {
  int e = blockIdx.x * blockDim.x + threadIdx.x;
  if (e < E) atomicAdd(&deg[col[e]], 1.0f);
}

__global__ void rsqrt_kernel(float* d, int n) {
  int i = blockIdx.x * blockDim.x + threadIdx.x;
  if (i < n) d[i] = rsqrtf(d[i]);  // deg >= 1 always
}

// ---------------- CSR build (counts -> scan -> bucket) ----------------

__global__ void count_kernel(const int* __restrict__ col, int* __restrict__ counts, int E) {
  int e = blockIdx.x * blockDim.x + threadIdx.x;
  if (e < E) atomicAdd(&counts[col[e]], 1);
}

// per-256-block exclusive scan, emits block totals
__global__ void scan_block_kernel(const int* __restrict__ in, int* __restrict__ out,
                                  int* __restrict__ bsum, int n) {
  __shared__ int tmp[256];
  int i = blockIdx.x * 256 + threadIdx.x;
  int v = (i < n) ? in[i] : 0;
  tmp[threadIdx.x] = v;
  __syncthreads();
  for (int off = 1; off < 256; off <<= 1) {
    int t = (threadIdx.x >= off) ? tmp[threadIdx.x - off] : 0;
    __syncthreads();
    tmp[threadIdx.x] += t;
    __syncthreads();
  }
  if (i < n) out[i] = tmp[threadIdx.x] - v;  // exclusive
  if (threadIdx.x == 255) bsum[blockIdx.x] = tmp[255];
}

// single-block exclusive scan of block totals (nb <= 512)
__global__ void scan_top_kernel(int* __restrict__ bsum, int nb) {
  __shared__ int tmp[512];
  int t = threadIdx.x;
  int v = (t < nb) ? bsum[t] : 0;
  tmp[t] = v;
  __syncthreads();
  for (int off = 1; off < 512; off <<= 1) {
    int a = (t >= off) ? tmp[t - off] : 0;
    __syncthreads();
    tmp[t] += a;
    __syncthreads();
  }
  if (t < nb) bsum[t] = tmp[t] - v;  // exclusive
}

__global__ void scan_add_kernel(int* __restrict__ out, const int* __restrict__ bsum, int n) {
  int i = blockIdx.x * 256 + threadIdx.x;
  if (i < n) out[i] += bsum[blockIdx.x];
}

// bucket edges by destination; weight = dis[row]*dis[col]
__global__ void build_csr_kernel(const int* __restrict__ row, const int* __restrict__ col,
                                 const float* __restrict__ dis, int* __restrict__ cursor,
                                 int* __restrict__ csrc, float* __restrict__ cwe, int E) {
  int e = blockIdx.x * blockDim.x + threadIdx.x;
  if (e >= E) return;
  int r = row[e], c = col[e];
  int pos = atomicAdd(&cursor[c], 1);
  csrc[pos] = r;
  cwe[pos] = dis[r] * dis[c];
}

// ---------------- fused APPNP hop: gather + self-loop + teleport ----------------
// out[v] = 0.9*( sum_j w_j * xin[src_j] + dis[v]^2 * xin[v] ) + 0.1*h[v]
// One wave per destination node; 128 feats as float4/lane; no atomics, no zero-fill.
__global__ void gather_kernel(const float* __restrict__ xin, const float* __restrict__ h,
                              const float* __restrict__ dis, const int* __restrict__ offsets,
                              const int* __restrict__ csrc, const float* __restrict__ cwe,
                              float* __restrict__ out, int N, int E) {
  int v = blockIdx.x * 8 + (threadIdx.x >> 5);
  if (v >= N) return;  // wave-uniform
  int lane = threadIdx.x & 31;
  int beg = offsets[v];
  int end = (v + 1 < N) ? offsets[v + 1] : E;
  float4 acc = make_float4(0.f, 0.f, 0.f, 0.f);
  for (int base = beg; base < end; base += 32) {
    int nume = end - base;
    if (nume > 32) nume = 32;
    int s = 0;
    float w = 0.0f;
    if (lane < nume) {
      s = csrc[base + lane];
      w = cwe[base + lane];
    }
    for (int t = 0; t < nume; ++t) {
      int ss = __shfl(s, t, 32);
      float ww = __shfl(w, t, 32);
      float4 xv = ((const float4*)(xin + (size_t)ss * FD))[lane];
      acc.x += ww * xv.x;
      acc.y += ww * xv.y;
      acc.z += ww * xv.z;
      acc.w += ww * xv.w;
    }
  }
  float d = dis[v];
  float sl = d * d;
  float4 cv = ((const float4*)(xin + (size_t)v * FD))[lane];
  float4 hv = ((const float4*)(h + (size_t)v * FD))[lane];
  float4 o;
  o.x = 0.9f * (acc.x + sl * cv.x) + 0.1f * hv.x;
  o.y = 0.9f * (acc.y + sl * cv.y) + 0.1f * hv.y;
  o.z = 0.9f * (acc.z + sl * cv.z) + 0.1f * hv.z;
  o.w = 0.9f * (acc.w + sl * cv.w) + 0.1f * hv.w;
  ((float4*)(out + (size_t)v * FD))[lane] = o;
}

// ---------------- moments: G += (mask*x)^T x ; Hm += (mask*x)^T y ----------------
// 8 waves/WG; wave w owns output feature-rows [w*16, w*16+16): 8 G tiles + 3 H tiles.
__global__ void moment_kernel(const float* __restrict__ x2, const float* __restrict__ mask,
                              const float* __restrict__ y, float* __restrict__ G,
                              float* __restrict__ Hm, int N, int npw) {
  int wave = threadIdx.x >> 5;
  int lane = threadIdx.x & 31;
  int l15 = lane & 15;
  int hi = lane >> 4;  // 0: K=0,1   1: K=2,3
  int m_base = wave * 16;
  int start = blockIdx.x * npw;
  int end = start + npw;
  if (end > N) end = N;

  v8f g[8], hacc[3];
#pragma unroll
  for (int t = 0; t < 8; ++t)
#pragma unroll
    for (int r = 0; r < 8; ++r) g[t][r] = 0.0f;
#pragma unroll
  for (int t = 0; t < 3; ++t)
#pragma unroll
    for (int r = 0; r < 8; ++r) hacc[t][r] = 0.0f;

  for (int k0 = start; k0 + 3 < end; k0 += 4) {
    int nA = k0 + hi * 2;
    const float* r0 = x2 + (size_t)nA * FD;
    const float* r1 = r0 + FD;
    float m0 = mask[nA];
    float m1 = mask[nA + 1];
    v2f a;  // A[m][k] = mask*x2[node k][feat m]
    a.x = r0[m_base + l15] * m0;
    a.y = r1[m_base + l15] * m1;
#pragma unroll
    for (int nt = 0; nt < 8; ++nt) {  // B[k][n] = x2[node k][feat n]
      v2f b;
      b.x = r0[nt * 16 + l15];
      b.y = r1[nt * 16 + l15];
      g[nt] = wmma4(a, b, g[nt]);
    }
#pragma unroll
    for (int ct = 0; ct < 3; ++ct) {  // B[k][n] = y[node k][class n] (pad 0)
      int c0 = ct * 16 + l15;
      v2f b;
      b.x = (c0 < NC) ? y[(size_t)nA * NC + c0] : 0.0f;
      b.y = (c0 < NC) ? y[(size_t)(nA + 1) * NC + c0] : 0.0f;
      hacc[ct] = wmma4(a, b, hacc[ct]);
    }
  }
  // C/D layout: VGPR r -> M = r (lanes 0-15) or 8+r (lanes 16-31); N = l15
#pragma unroll
  for (int nt = 0; nt < 8; ++nt)
#pragma unroll
    for (int r = 0; r < 8; ++r)
      atomicAdd(&G[(m_base + r + hi * 8) * FD + nt * 16 + l15], g[nt][r]);
#pragma unroll
  for (int ct = 0; ct < 3; ++ct)
#pragma unroll
    for (int r = 0; r < 8; ++r)
      atomicAdd(&Hm[(m_base + r + hi * 8) * NCP + ct * 16 + l15], hacc[ct][r]);
}

// ---------------- ridge solve: Gauss-Jordan on [G + REG*I | Hm] in LDS ----------------
// 128 x 176 fp32 = 90KB of the 320KB WGP LDS. G is SPD -> no pivoting needed.
__global__ void solve_kernel(const float* __restrict__ G, const float* __restrict__ Hm,
                             float* __restrict__ S) {
  __shared__ float M[FD][FD + NCP];
  int t = threadIdx.x;  // 256 threads
  for (int idx = t; idx < FD * FD; idx += 256) {
    int r = idx >> 7, c = idx & 127;
    float v = G[idx];
    if (r == c) v += 1e-5f;
    M[r][c] = v;
  }
  for (int idx = t; idx < FD * NCP; idx += 256) {
    int r = idx / NCP, c = idx - r * NCP;
    M[r][FD + c] = Hm[idx];
  }
  __syncthreads();
  int rr = t >> 1;   // row owned by this thread
  int half = t & 1;  // column half [half*88, half*88+88)
  for (int p = 0; p < FD; ++p) {
    float inv = 1.0f / M[p][p];
    __syncthreads();
    for (int c = t; c < FD + NCP; c += 256) M[p][c] *= inv;
    __syncthreads();
    float f = (rr != p) ? M[rr][p] : 0.0f;
    __syncthreads();
    int cb = half * 88;
    for (int c = cb; c < cb + 88; ++c) M[rr][c] -= f * M[p][c];
    __syncthreads();
  }
  for (int idx = t; idx < FD * NCP; idx += 256) {
    int r = idx / NCP, c = idx - r * NCP;
    S[idx] = M[r][FD + c];
  }
}

// ---------------- out(N x 40) = x2(N x 128) @ S(128 x 48pad) ----------------
// S staged into LDS (async-to-LDS when available); one wave per 16-node tile.
__global__ void gemm_kernel(const float* __restrict__ x2, const float* __restrict__ S,
                            float* __restrict__ out, int N) {
  __shared__ float Sl[FD * NCP];  // 24KB
#if __has_builtin(__builtin_amdgcn_global_load_async_to_lds_b128)
  for (int i = threadIdx.x; i < FD * NCP / 4; i += 256) {
    __builtin_amdgcn_global_load_async_to_lds_b128(
        (__attribute__((address_space(1))) v4i*)(S + i * 4),
        (__attribute__((address_space(3))) v4i*)(Sl + i * 4), 0, 0);
  }
#if __has_builtin(__builtin_amdgcn_s_wait_asynccnt)
  __builtin_amdgcn_s_wait_asynccnt(0);
#else
  asm volatile("s_wait_asynccnt 0" ::: "memory");
#endif
#else
  for (int i = threadIdx.x; i < FD * NCP; i += 256) Sl[i] = S[i];
#endif
  __syncthreads();

  int gwid = blockIdx.x * 8 + (threadIdx.x >> 5);
  int node_base = gwid * 16;
  if (node_base >= N) return;  // wave-uniform
  int lane = threadIdx.x & 31;
  int l15 = lane & 15;
  int hi2 = (lane >> 4) * 2;
  const float* arow = x2 + (size_t)(node_base + l15) * FD + hi2;

  v8f acc[3];
#pragma unroll
  for (int t = 0; t < 3; ++t)
#pragma unroll
    for (int r = 0; r < 8; ++r) acc[t][r] = 0.0f;

  for (int k0 = 0; k0 < FD; k0 += 4) {
    v2f a = *(const v2f*)(arow + k0);  // A[m][k] = x2[node][k], contiguous pair
#pragma unroll
    for (int ct = 0; ct < 3; ++ct) {  // B[k][n] = S[k][class] from LDS
      v2f b;
      b.x = Sl[(k0 + hi2) * NCP + ct * 16 + l15];
      b.y = Sl[(k0 + hi2 + 1) * NCP + ct * 16 + l15];
      acc[ct] = wmma4(a, b, acc[ct]);
    }
  }
#pragma unroll
  for (int ct = 0; ct < 3; ++ct) {
    int colc = ct * 16 + l15;
    if (colc < NC) {
#pragma unroll
      for (int r = 0; r < 8; ++r) {
        int node = node_base + r + (lane >> 4) * 8;
        out[(size_t)node * NC + colc] = acc[ct][r];
      }
    }
  }
}

// ---------------- host-side orchestration ----------------

extern "C" void kernel_launch(void* const* d_in, const int* in_sizes, int n_in,
                              void* d_out, int out_size, void* d_ws, size_t ws_size,
                              hipStream_t stream) {
  const float* x    = (const float*)d_in[0];
  const int*   ei   = (const int*)d_in[1];
  const float* y    = (const float*)d_in[2];
  const float* mask = (const float*)d_in[3];
  float* out = (float*)d_out;
  int N = in_sizes[0] / FD;
  int E = in_sizes[1] / 2;
  const int* row = ei;
  const int* col = ei + E;
  int NBLK = (N + 255) / 256;  // <= 512 for this problem size

  char* ws = (char*)d_ws;
  size_t off = 0;
  float* dis     = (float*)(ws + off); off += (size_t)N * 4;
  int*   counts  = (int*)(ws + off);   off += (size_t)N * 4;
  int*   offsets = (int*)(ws + off);   off += (size_t)N * 4;
  int*   cursor  = (int*)(ws + off);   off += (size_t)N * 4;
  int*   bsum    = (int*)(ws + off);   off += (size_t)512 * 4;
  int*   csrc    = (int*)(ws + off);   off += (size_t)E * 4;
  float* cwe     = (float*)(ws + off); off += (size_t)E * 4;
  float* buf0    = (float*)(ws + off); off += (size_t)N * FD * 4;
  float* buf1    = (float*)(ws + off); off += (size_t)N * FD * 4;
  float* G       = (float*)(ws + off); off += (size_t)FD * FD * 4;
  float* Hm      = (float*)(ws + off); off += (size_t)FD * NCP * 4;
  float* S       = (float*)(ws + off); off += (size_t)FD * NCP * 4;

  dim3 b256(256);
  int gN = (N + 255) / 256;
  int gE = (E + 255) / 256;

  // gcn_norm: deg (with self loop), dis = rsqrt(deg)
  fill_kernel<<<gN, b256, 0, stream>>>(dis, 1.0f, N);
  deg_kernel<<<gE, b256, 0, stream>>>(col, dis, E);
  rsqrt_kernel<<<gN, b256, 0, stream>>>(dis, N);

  // CSR by destination: counts -> exclusive scan -> bucket edges
  fill_int_kernel<<<gN, b256, 0, stream>>>(counts, 0, N);
  count_kernel<<<gE, b256, 0, stream>>>(col, counts, E);
  scan_block_kernel<<<NBLK, b256, 0, stream>>>(counts, offsets, bsum, N);
  scan_top_kernel<<<1, dim3(512), 0, stream>>>(bsum, NBLK);
  scan_add_kernel<<<NBLK, b256, 0, stream>>>(offsets, bsum, N);
  copy_int_kernel<<<gN, b256, 0, stream>>>(offsets, cursor, N);
  build_csr_kernel<<<gE, b256, 0, stream>>>(row, col, dis, cursor, csrc, cwe, E);

  // APPNP hops (fused gather + self-loop + teleport; no atomics)
  int gG = (N + 7) / 8;
  gather_kernel<<<gG, b256, 0, stream>>>(x, x, dis, offsets, csrc, cwe, buf0, N, E);
  gather_kernel<<<gG, b256, 0, stream>>>(buf0, x, dis, offsets, csrc, cwe, buf1, N, E);

  // moments via fp32 WMMA
  fill_kernel<<<(FD * FD + 255) / 256, b256, 0, stream>>>(G, 0.0f, FD * FD);
  fill_kernel<<<(FD * NCP + 255) / 256, b256, 0, stream>>>(Hm, 0.0f, FD * NCP);
  int NB = 240;
  int npw = (((N + NB - 1) / NB) + 3) & ~3;  // nodes per WG, multiple of 4
  moment_kernel<<<NB, b256, 0, stream>>>(buf1, mask, y, G, Hm, N, npw);

  // ridge solve (single workgroup, LDS Gauss-Jordan)
  solve_kernel<<<1, b256, 0, stream>>>(G, Hm, S);

  // out = buf1 @ S via fp32 WMMA with LDS-staged S
  int tiles = (N + 15) / 16;
  gemm_kernel<<<(tiles + 7) / 8, b256, 0, stream>>>(buf1, S, out, N);

  (void)n_in; (void)out_size; (void)ws_size;
}